// QCNN_57921928954275
// MI455X (gfx1250) — compile-verified
//
#include <hip/hip_runtime.h>
#include <hip/hip_bf16.h>
#include <math.h>

// ---------------------------------------------------------------------------
// QCNN (8 qubits, depth 2) reduced to  z_b = psi_b^T A psi_b  where
//   psi_b = real product state from RY-embedding of x[b],
//   A     = Re(U^H D U),  U = fixed circuit unitary, D = diag(+-1 on qubit 0).
// Kernel 1 builds U column-by-column (statevector sim in LDS).
// Kernel 2 builds A.
// Kernel 3: B x 256 x 256 f32 GEMM via V_WMMA_F32_16X16X4_F32; each wave owns
//   TWO 16-batch tiles sharing one B operand stream -> two independent WMMA
//   accumulation chains (hides XDL latency) and half the L2 traffic.
// ---------------------------------------------------------------------------

typedef float v2f __attribute__((ext_vector_type(2)));
typedef float v8f __attribute__((ext_vector_type(8)));

#define NQ 8
#define DIM 256           // 2^NQ
#define PSTRIDE 260       // 256 + 4 pad: row m -> bank offset 4m (conflict-free)

// ---------------------------------------------------------------------------
// Kernel 1: build U. Block c computes column c of U (circuit applied to |c>).
// 128 threads; statevector (256 complex) in LDS. Each gate partitions the 256
// amplitudes into 128 thread-exclusive pairs -> barrier only between gates.
// Qubit q lives at bit position (7 - q) of the flat amplitude index.
// ---------------------------------------------------------------------------
__global__ void __launch_bounds__(128) build_U_kernel(
    const float* __restrict__ conv,   // [2][8][6]
    const float* __restrict__ pool,   // [2][4][3]
    float* __restrict__ Ur,           // [256][256] row = amplitude k, col = c
    float* __restrict__ Ui) {
  __shared__ float sr[DIM], si[DIM];
  const int c = blockIdx.x;
  const int t = threadIdx.x;

  sr[t]       = (t == c)       ? 1.f : 0.f;  si[t]       = 0.f;
  sr[t + 128] = (t + 128 == c) ? 1.f : 0.f;  si[t + 128] = 0.f;
  __syncthreads();

  auto apply = [&](int tq, int cq,
                   float u00r, float u00i, float u01r, float u01i,
                   float u10r, float u10i, float u11r, float u11i) {
    const int bp   = 7 - tq;
    const int low  = t & ((1 << bp) - 1);
    const int high = (t >> bp) << (bp + 1);
    const int a0   = high | low;
    const int a1   = a0 | (1 << bp);
    bool act = true;
    if (cq >= 0) act = ((a0 >> (7 - cq)) & 1) != 0;
    if (act) {
      float r0 = sr[a0], i0 = si[a0], r1 = sr[a1], i1 = si[a1];
      float n0r = u00r * r0 - u00i * i0 + u01r * r1 - u01i * i1;
      float n0i = u00r * i0 + u00i * r0 + u01r * i1 + u01i * r1;
      float n1r = u10r * r0 - u10i * i0 + u11r * r1 - u11i * i1;
      float n1i = u10r * i0 + u10i * r0 + u11r * i1 + u11i * r1;
      sr[a0] = n0r; si[a0] = n0i; sr[a1] = n1r; si[a1] = n1i;
    }
    __syncthreads();
  };

  auto cry = [&](float th, int cq, int tq) {
    float ss, cc; __sincosf(th * 0.5f, &ss, &cc);
    apply(tq, cq, cc, 0.f, -ss, 0.f, ss, 0.f, cc, 0.f);
  };
  auto crx = [&](float th, int cq, int tq) {
    float ss, cc; __sincosf(th * 0.5f, &ss, &cc);
    apply(tq, cq, cc, 0.f, 0.f, -ss, 0.f, -ss, cc, 0.f);
  };
  auto ry1 = [&](float th, int tq) {
    float ss, cc; __sincosf(th * 0.5f, &ss, &cc);
    apply(tq, -1, cc, 0.f, -ss, 0.f, ss, 0.f, cc, 0.f);
  };
  auto cnot = [&](int cq, int tq) {
    apply(tq, cq, 0.f, 0.f, 1.f, 0.f, 1.f, 0.f, 0.f, 0.f);
  };
  auto cu3 = [&](const float* p, int cq, int tq) {
    float th = p[0], ph = p[1], la = p[2];
    float st, ct;   __sincosf(th * 0.5f, &st, &ct);
    float sl, cl;   __sincosf(la, &sl, &cl);
    float sp, cp;   __sincosf(ph, &sp, &cp);
    float spl, cpl; __sincosf(ph + la, &spl, &cpl);
    apply(tq, cq, ct, 0.f, -cl * st, -sl * st, cp * st, sp * st, cpl * ct, spl * ct);
  };

  auto conv_block = [&](const float* w, int a, int b) {
    cry(w[0], a, b); cry(w[1], b, a);
    crx(w[2], a, b); crx(w[3], b, a);
    cnot(a, b);
    ry1(w[4], a); ry1(w[5], b);
  };

  // ----- layer 0 conv: pairs (0,1)(2,3)(4,5)(6,7), idx 0..3 -----
  conv_block(conv + 0 * 6, 0, 1);
  conv_block(conv + 1 * 6, 2, 3);
  conv_block(conv + 2 * 6, 4, 5);
  conv_block(conv + 3 * 6, 6, 7);
  // ----- layer 0 pool: ctrl=1->tgt=0, 3->2, 5->4, 7->6 -----
  cu3(pool + 0, 1, 0);
  cu3(pool + 3, 3, 2);
  cu3(pool + 6, 5, 4);
  cu3(pool + 9, 7, 6);
  // ----- layer 1 conv: pairs (0,2)(4,6), idx 0..1 of layer 1 -----
  conv_block(conv + 48 + 0 * 6, 0, 2);
  conv_block(conv + 48 + 1 * 6, 4, 6);
  // ----- layer 1 pool: ctrl=2->tgt=0, 6->4 -----
  cu3(pool + 12 + 0, 2, 0);
  cu3(pool + 12 + 3, 6, 4);

  Ur[t * DIM + c]         = sr[t];       Ui[t * DIM + c]         = si[t];
  Ur[(t + 128) * DIM + c] = sr[t + 128]; Ui[(t + 128) * DIM + c] = si[t + 128];
}

// ---------------------------------------------------------------------------
// Kernel 2: A[i][j] = sum_k d_k (Ur[k][i]Ur[k][j] + Ui[k][i]Ui[k][j]),
//           d_k = +1 if bit7(k)==0 else -1 (PauliZ on qubit 0 = MSB).
// ---------------------------------------------------------------------------
__global__ void __launch_bounds__(256) build_A_kernel(
    const float* __restrict__ Ur, const float* __restrict__ Ui,
    float* __restrict__ Amat) {
  const int i = blockIdx.x;
  const int j = threadIdx.x;
  float acc = 0.f;
  for (int k = 0; k < DIM; ++k) {
    float d = (k & 128) ? -1.f : 1.f;
    acc += d * (Ur[k * DIM + i] * Ur[k * DIM + j] +
                Ui[k * DIM + i] * Ui[k * DIM + j]);
  }
  Amat[i * DIM + j] = acc;
}

// ---------------------------------------------------------------------------
// Kernel 3: single-wave workgroups; each wave owns 32 batch rows = two 16-row
// WMMA tiles sharing one B-operand stream.
// WMMA operand layouts (wave32, 32-bit):
//   A op: lane l (l<16): M=l,    VGPR0=K0+0, VGPR1=K0+1
//         lane l (>=16): M=l-16, VGPR0=K0+2, VGPR1=K0+3
//   B op: lane l (l<16): N=l,    VGPR0=row K0+0, VGPR1=row K0+1
//         lane l (>=16): N=l-16, VGPR0=row K0+2, VGPR1=row K0+3
//   D:    lane l, VGPR r: M = r + 8*(l>=16), N = l&15
// ---------------------------------------------------------------------------
__global__ void __launch_bounds__(32) qcnn_gemm_kernel(
    const float* __restrict__ x,      // [B][8]
    const float* __restrict__ Amat,   // [256][256], symmetric
    float* __restrict__ out) {        // [B]
  __shared__ float psi[32 * PSTRIDE];   // 33.3 KB
  const int lane = threadIdx.x & 31;
  const int b0   = blockIdx.x * 32;
  float* P = psi;

  const int m    = lane & 15;   // batch row within tile / N index
  const int half = lane >> 4;   // lane half

  // ---- build psi: lane (m,half) fills halves of rows m and m+16 ----
#pragma unroll
  for (int rr = 0; rr < 2; ++rr) {
    const int row = m + rr * 16;
    const float* xr = x + (size_t)(b0 + row) * NQ;
    float cq[NQ], sq[NQ];
#pragma unroll
    for (int q = 0; q < NQ; ++q) {
      float s, cc; __sincosf(xr[q] * 0.5f, &s, &cc);
      cq[q] = cc; sq[q] = s;
    }
    float base = half ? sq[0] : cq[0];   // qubit 0 = bit 7 = our half bit
    float* Prow = P + row * PSTRIDE + half * 128;
#pragma unroll 8
    for (int a = 0; a < 128; ++a) {
      float p = base;
      p *= (a & 64) ? sq[1] : cq[1];
      p *= (a & 32) ? sq[2] : cq[2];
      p *= (a & 16) ? sq[3] : cq[3];
      p *= (a & 8)  ? sq[4] : cq[4];
      p *= (a & 4)  ? sq[5] : cq[5];
      p *= (a & 2)  ? sq[6] : cq[6];
      p *= (a & 1)  ? sq[7] : cq[7];
      Prow[a] = p;
    }
  }
  __syncthreads();   // single-wave workgroup: lowers to waits, no real barrier

  const int koff = half * 2;
  const float* Aop0 = P + m * PSTRIDE + koff;          // psi row m
  const float* Aop1 = P + (m + 16) * PSTRIDE + koff;   // psi row m+16
  float acc0[8] = {0.f, 0.f, 0.f, 0.f, 0.f, 0.f, 0.f, 0.f};
  float acc1[8] = {0.f, 0.f, 0.f, 0.f, 0.f, 0.f, 0.f, 0.f};

  for (int i0 = 0; i0 < DIM; i0 += 16) {
    v8f c0 = {0.f, 0.f, 0.f, 0.f, 0.f, 0.f, 0.f, 0.f};
    v8f c1 = {0.f, 0.f, 0.f, 0.f, 0.f, 0.f, 0.f, 0.f};
    // B operand row K0+k, col i0+n  ==  Amat[i0+n][K0+k] (A symmetric):
    // each lane streams contiguous K from row (i0+m); shared by both tiles.
    const float* Bop = Amat + (size_t)(i0 + m) * DIM + koff;
#pragma unroll
    for (int k0 = 0; k0 < DIM; k0 += 4) {
      v2f bv = *(const v2f*)(Bop + k0);
      v2f a0 = *(const v2f*)(Aop0 + k0);
      v2f a1 = *(const v2f*)(Aop1 + k0);
      c0 = __builtin_amdgcn_wmma_f32_16x16x4_f32(
          false, a0, false, bv, (short)0, c0, false, false);
      c1 = __builtin_amdgcn_wmma_f32_16x16x4_f32(
          false, a1, false, bv, (short)0, c1, false, false);
    }
    // z[b + r + 8*half] += Y[r][n] * psi[row][i0 + n]
    const float* Pcol0 = P + (half * 8) * PSTRIDE + i0 + m;
    const float* Pcol1 = P + (16 + half * 8) * PSTRIDE + i0 + m;
#pragma unroll
    for (int r = 0; r < 8; ++r) {
      acc0[r] += c0[r] * Pcol0[r * PSTRIDE];
      acc1[r] += c1[r] * Pcol1[r * PSTRIDE];
    }
  }

  // reduce the 16 lanes of each half (xor masks stay within a half)
#pragma unroll
  for (int r = 0; r < 8; ++r) {
    float v0 = acc0[r];
    v0 += __shfl_xor(v0, 1, 32);
    v0 += __shfl_xor(v0, 2, 32);
    v0 += __shfl_xor(v0, 4, 32);
    v0 += __shfl_xor(v0, 8, 32);
    acc0[r] = v0;
    float v1 = acc1[r];
    v1 += __shfl_xor(v1, 1, 32);
    v1 += __shfl_xor(v1, 2, 32);
    v1 += __shfl_xor(v1, 4, 32);
    v1 += __shfl_xor(v1, 8, 32);
    acc1[r] = v1;
  }
  if (m == 0) {
#pragma unroll
    for (int r = 0; r < 8; ++r) {
      out[b0 + half * 8 + r]      = acc0[r];
      out[b0 + 16 + half * 8 + r] = acc1[r];
    }
  }
}

// ---------------------------------------------------------------------------
extern "C" void kernel_launch(void* const* d_in, const int* in_sizes, int n_in,
                              void* d_out, int out_size, void* d_ws, size_t ws_size,
                              hipStream_t stream) {
  const float* conv = (const float*)d_in[0];   // [2][8][6]
  const float* pool = (const float*)d_in[1];   // [2][4][3]
  const float* x    = (const float*)d_in[2];   // [B][8]
  float* out = (float*)d_out;

  float* Ur   = (float*)d_ws;                  // 256*256 f32
  float* Ui   = Ur + DIM * DIM;                // 256*256 f32
  float* Amat = Ui + DIM * DIM;                // 256*256 f32  (total 768 KB)

  const int B = in_sizes[2] / NQ;              // 131072
  const int blocks = B / 32;                   // 4096 (two 16-row tiles/wave)

  build_U_kernel<<<DIM, 128, 0, stream>>>(conv, pool, Ur, Ui);
  build_A_kernel<<<DIM, 256, 0, stream>>>(Ur, Ui, Amat);
  qcnn_gemm_kernel<<<blocks, 32, 0, stream>>>(x, Amat, out);
}